// Backbone2_48928267436129
// MI455X (gfx1250) — compile-verified
//
#include <hip/hip_runtime.h>
#include <math.h>

typedef __attribute__((ext_vector_type(2))) float v2f;
typedef __attribute__((ext_vector_type(8))) float v8f;

#define GF_BT    1
#define GF_RELU  2
#define GF_RELUA 4

// ---------------------------------------------------------------- GEMM (WMMA f32 16x16x4)
struct GemmP {
  const float* A;  long long ao, ai; int lda;
  const float* A2;                      // optional second A (elementwise add), same indexing
  const float* Bm; long long bo, bi; int ldb;
  const float* bias;                    // per output column, optional
  const float* Res; int ldr;            // optional residual (row-major)
  float* C;        long long co, ci; int ldc;
  int M, N, K, hdim, flags;
};

// MODE: 0 = plain NN, 1 = B transposed (NT), 2 = ReLU applied to A, 3 = dual-A (A + A2)
template<int NT, int MODE>
__global__ __launch_bounds__(128)
void k_gemm(GemmP g) {
  constexpr bool BT  = (MODE == 1);
  constexpr bool RA  = (MODE == 2);
  constexpr bool A2E = (MODE == 3);
  const int wave = threadIdx.x >> 5;
  const int lane = threadIdx.x & 31;
  const int l16  = lane & 15;
  const int hi2  = (lane >> 4) * 2;     // 0 for lanes 0-15, 2 for 16-31
  const int z  = blockIdx.z;
  const int zb = z / g.hdim, zi = z % g.hdim;
  const float* A  = g.A + (long long)zb * g.ao + (long long)zi * g.ai;
  const float* Bm = g.Bm + (long long)zb * g.bo + (long long)zi * g.bi;
  float* C = g.C + (long long)zb * g.co + (long long)zi * g.ci;
  const int m0 = blockIdx.y * 64 + wave * 16;
  const int n0 = blockIdx.x * (16 * NT);
  const int row = m0 + l16;
  const bool rok = row < g.M;
  const int rowc = rok ? row : (g.M - 1);          // clamp: always-legal address
  const float* ap = A + (long long)rowc * g.lda + hi2;
  const float* a2p = nullptr;
  if (A2E) a2p = g.A2 + (long long)zb * g.ao + (long long)zi * g.ai
                      + (long long)rowc * g.lda + hi2;

  const float* bp[NT]; bool bok[NT];
#pragma unroll
  for (int t = 0; t < NT; ++t) {
    int n = n0 + t * 16 + l16;
    bok[t] = n < g.N;
    int nc = bok[t] ? n : (g.N - 1);               // clamp: always-legal address
    bp[t] = BT ? (Bm + (long long)nc * g.ldb + hi2)
               : (Bm + (long long)hi2 * g.ldb + nc);
  }
  const long long bstep = BT ? 4 : 4 * (long long)g.ldb;

  v8f acc[NT];
#pragma unroll
  for (int t = 0; t < NT; ++t)
#pragma unroll
    for (int e = 0; e < 8; ++e) acc[t][e] = 0.f;

#pragma unroll 2
  for (int k = 0; k < g.K; k += 4) {
    // A fragment: VGPR0 = A[row, k+hi2], VGPR1 = A[row, k+hi2+1]  (branchless)
    float a0 = ap[0], a1 = ap[1];
    ap += 4;
    if (A2E) { a0 += a2p[0]; a1 += a2p[1]; a2p += 4; }
    if (!rok) { a0 = 0.f; a1 = 0.f; }
    if (RA) { a0 = fmaxf(a0, 0.f); a1 = fmaxf(a1, 0.f); }
    v2f af; af.x = a0; af.y = a1;

    float b0[NT], b1[NT];
#pragma unroll
    for (int t = 0; t < NT; ++t) {
      b0[t] = bp[t][0];
      b1[t] = BT ? bp[t][1] : bp[t][g.ldb];
      bp[t] += bstep;
      if (!bok[t]) { b0[t] = 0.f; b1[t] = 0.f; }
    }
#pragma unroll
    for (int t = 0; t < NT; ++t) {
      v2f bf; bf.x = b0[t]; bf.y = b1[t];
      acc[t] = __builtin_amdgcn_wmma_f32_16x16x4_f32(false, af, false, bf, (short)0,
                                                     acc[t], false, false);
    }
  }

#pragma unroll
  for (int t = 0; t < NT; ++t) {
#pragma unroll
    for (int r = 0; r < 8; ++r) {
      const int mr = m0 + r + 4 * hi2;             // D layout: VGPR r = rows r / r+8
      const int nc = n0 + t * 16 + l16;
      if (mr < g.M && nc < g.N) {
        float v = acc[t][r];
        if (g.bias) v += g.bias[nc];
        if (g.Res)  v += g.Res[(long long)mr * g.ldr + nc];
        if (g.flags & GF_RELU) v = fmaxf(v, 0.f);
        C[(long long)mr * g.ldc + nc] = v;
      }
    }
  }
}

// ---------------------------------------------------------------- LayerNorm
__device__ __forceinline__ float mask_scale(const float* mask, int mode, long long row, int Np) {
  if (mode == 0 || !mask) return 1.f;
  if (mode == 1) return mask[row];
  long long NN = (long long)Np * Np;
  long long b = row / NN, r = row - b * NN;
  long long i = r / Np, j = r - i * Np;
  return mask[b * Np + i] * mask[b * Np + j];
}

__global__ __launch_bounds__(128)
void k_ln(const float* x, const float* res, const float* gw, const float* bw,
          const float* mask, int premode, int postmode, int Np,
          float* out, int D) {
  const long long row = blockIdx.x;
  const float pre  = mask_scale(mask, premode, row, Np);
  const float post = mask_scale(mask, postmode, row, Np);
  const float* xr = x + row * (long long)D;
  const float* rr = res ? res + row * (long long)D : nullptr;
  float v[4]; int cnt = 0; float s = 0.f;
  for (int c = threadIdx.x; c < D; c += 128) {
    float t = xr[c] * pre + (rr ? rr[c] : 0.f);
    v[cnt++] = t; s += t;
  }
  __shared__ float sh[128];
  sh[threadIdx.x] = s; __syncthreads();
  for (int o = 64; o > 0; o >>= 1) { if (threadIdx.x < o) sh[threadIdx.x] += sh[threadIdx.x + o]; __syncthreads(); }
  const float mean = sh[0] / D; __syncthreads();
  float vs = 0.f;
  for (int q = 0; q < cnt; ++q) { float d = v[q] - mean; vs += d * d; }
  sh[threadIdx.x] = vs; __syncthreads();
  for (int o = 64; o > 0; o >>= 1) { if (threadIdx.x < o) sh[threadIdx.x] += sh[threadIdx.x + o]; __syncthreads(); }
  const float inv = rsqrtf(sh[0] / D + 1e-5f);
  cnt = 0;
  for (int c = threadIdx.x; c < D; c += 128)
    out[row * (long long)D + c] = ((v[cnt++] - mean) * inv * gw[c] + bw[c]) * post;
}

// ---------------------------------------------------------------- Softmax kernels
__global__ __launch_bounds__(128)
void k_softmax_ipa(float* a, const float* ptqk, const float* bproj,
                   const float* qsq, const float* ksq, const float* head_w,
                   const float* mask, int Nn, int Hh) {
  const int i = blockIdx.x, h = blockIdx.y, b = blockIdx.z;
  const float hw = log1pf(expf(head_w[h])) * 0.13608276348795434f;   // softplus * sqrt(1/54)
  const float s1 = 0.14433756729740643f;                             // sqrt(1/48)
  const float s2 = 0.57735026918962584f;                             // sqrt(1/3)
  const float mi = mask[b * Nn + i];
  const float qs = qsq[((long long)b * Nn + i) * Hh + h];
  const long long rb   = (((long long)b * Hh + h) * Nn + i) * (long long)Nn;
  const long long prow = ((long long)b * Nn + i) * (long long)Nn;
  float vals[4]; int cnt = 0; float mx = -3.0e38f;
  for (int j = threadIdx.x; j < Nn; j += 128) {
    float l = a[rb + j] * s1 + s2 * bproj[(prow + j) * Hh + h]
            - 0.5f * hw * (qs + ksq[((long long)b * Nn + j) * Hh + h] - 2.f * ptqk[rb + j])
            + 100000.f * (mi * mask[b * Nn + j] - 1.f);
    vals[cnt++] = l; mx = fmaxf(mx, l);
  }
  __shared__ float sh[128];
  sh[threadIdx.x] = mx; __syncthreads();
  for (int o = 64; o > 0; o >>= 1) { if (threadIdx.x < o) sh[threadIdx.x] = fmaxf(sh[threadIdx.x], sh[threadIdx.x + o]); __syncthreads(); }
  mx = sh[0]; __syncthreads();
  float s = 0.f;
  for (int q = 0; q < cnt; ++q) { vals[q] = expf(vals[q] - mx); s += vals[q]; }
  sh[threadIdx.x] = s; __syncthreads();
  for (int o = 64; o > 0; o >>= 1) { if (threadIdx.x < o) sh[threadIdx.x] += sh[threadIdx.x + o]; __syncthreads(); }
  const float inv = 1.f / sh[0];
  cnt = 0;
  for (int j = threadIdx.x; j < Nn; j += 128) a[rb + j] = vals[cnt++] * inv;
}

__global__ __launch_bounds__(128)
void k_softmax_tfmr(float* a, const float* mask, int Nn, int Hh) {
  const int i = blockIdx.x, h = blockIdx.y, b = blockIdx.z;
  const long long rb = (((long long)b * Hh + h) * Nn + i) * (long long)Nn;
  const float sc = 0.11180339887498949f;   // 1/sqrt(80)
  float vals[4]; int cnt = 0; float mx = -3.0e38f;
  for (int j = threadIdx.x; j < Nn; j += 128) {
    float l = a[rb + j] * sc + (1.f - mask[b * Nn + j]);
    vals[cnt++] = l; mx = fmaxf(mx, l);
  }
  __shared__ float sh[128];
  sh[threadIdx.x] = mx; __syncthreads();
  for (int o = 64; o > 0; o >>= 1) { if (threadIdx.x < o) sh[threadIdx.x] = fmaxf(sh[threadIdx.x], sh[threadIdx.x + o]); __syncthreads(); }
  mx = sh[0]; __syncthreads();
  float s = 0.f;
  for (int q = 0; q < cnt; ++q) { vals[q] = expf(vals[q] - mx); s += vals[q]; }
  sh[threadIdx.x] = s; __syncthreads();
  for (int o = 64; o > 0; o >>= 1) { if (threadIdx.x < o) sh[threadIdx.x] += sh[threadIdx.x + o]; __syncthreads(); }
  const float inv = 1.f / sh[0];
  cnt = 0;
  for (int j = threadIdx.x; j < Nn; j += 128) a[rb + j] = vals[cnt++] * inv;
}

// ---------------------------------------------------------------- small elementwise kernels
__global__ void k_maskmul(float* dst, const float* src, const float* mask, int D, long long total) {
  long long g = (long long)blockIdx.x * blockDim.x + threadIdx.x;
  if (g >= total) return;
  dst[g] = src[g] * mask[g / D];
}

__global__ void k_copy2d(float* dst, int dld, const float* src, int sld, int cols, long long total) {
  long long g = (long long)blockIdx.x * blockDim.x + threadIdx.x;
  if (g >= total) return;
  long long r = g / cols; int c = (int)(g - r * cols);
  dst[r * (long long)dld + c] = src[r * (long long)sld + c];
}

__global__ void k_edge_concat(float* E, const float* pair, const float* nbv, int Nn, long long total) {
  long long g = (long long)blockIdx.x * blockDim.x + threadIdx.x;
  if (g >= total) return;
  long long row = g / 384; int c = (int)(g - row * 384);
  long long NN = (long long)Nn * Nn;
  long long b = row / NN, r2 = row - b * NN;
  long long i = r2 / Nn, j = r2 - i * Nn;
  float v;
  if (c < 128)      v = pair[row * 128 + c];
  else if (c < 256) v = nbv[(b * Nn + i) * 128 + (c - 128)];
  else              v = nbv[(b * Nn + j) * 128 + (c - 256)];
  E[g] = v;
}

__global__ void k_rigid(float* outp, const float* inp, const float* R, const float* t, int P, long long total) {
  long long g = (long long)blockIdx.x * blockDim.x + threadIdx.x;
  if (g >= total) return;
  long long row = g / P;
  const float* v = inp + g * 3;
  const float* Rr = R + row * 9;
  const float* tr = t + row * 3;
  float x = v[0], y = v[1], z = v[2];
  float* o = outp + g * 3;
  o[0] = Rr[0]*x + Rr[1]*y + Rr[2]*z + tr[0];
  o[1] = Rr[3]*x + Rr[4]*y + Rr[5]*z + tr[1];
  o[2] = Rr[6]*x + Rr[7]*y + Rr[8]*z + tr[2];
}

__global__ void k_sumsq(float* out, const float* x, int ld, int K, long long rows) {
  long long g = (long long)blockIdx.x * blockDim.x + threadIdx.x;
  if (g >= rows) return;
  const float* p = x + g * ld;
  float s = 0.f;
  for (int k = 0; k < K; ++k) s += p[k] * p[k];
  out[g] = s;
}

__global__ void k_ipa_finalize(float* cat, const float* optb, const float* R, const float* t,
                               int Nn, int Hh, int Pv, long long total) {
  long long g = (long long)blockIdx.x * blockDim.x + threadIdx.x;
  if (g >= total) return;
  int p = (int)(g % Pv); long long g2 = g / Pv;
  int h = (int)(g2 % Hh); long long row = g2 / Hh;
  long long b = row / Nn; long long i = row - b * Nn;
  long long base = (((b * Hh + h) * Nn) + i) * (long long)(3 * Pv) + (long long)p * 3;
  const float* tr = t + row * 3;
  float vx = optb[base]     - tr[0];
  float vy = optb[base + 1] - tr[1];
  float vz = optb[base + 2] - tr[2];
  const float* Rr = R + row * 9;
  float lx = Rr[0]*vx + Rr[3]*vy + Rr[6]*vz;   // R^T * v
  float ly = Rr[1]*vx + Rr[4]*vy + Rr[7]*vz;
  float lz = Rr[2]*vx + Rr[5]*vy + Rr[8]*vz;
  float* cr = cat + row * 2112;
  cr[192 + h * 3 * Pv + p * 3 + 0] = lx;
  cr[192 + h * 3 * Pv + p * 3 + 1] = ly;
  cr[192 + h * 3 * Pv + p * 3 + 2] = lz;
  cr[480 + h * Pv + p] = sqrtf(lx*lx + ly*ly + lz*lz + 1e-8f);
}

// ---------------------------------------------------------------- quaternions / frames
__device__ __forceinline__ void quat2rot(float w, float x, float y, float z, float* R) {
  R[0] = 1.f - 2.f*(y*y + z*z); R[1] = 2.f*(x*y - w*z); R[2] = 2.f*(x*z + w*y);
  R[3] = 2.f*(x*y + w*z); R[4] = 1.f - 2.f*(x*x + z*z); R[5] = 2.f*(y*z - w*x);
  R[6] = 2.f*(x*z - w*y); R[7] = 2.f*(y*z + w*x); R[8] = 1.f - 2.f*(x*x + y*y);
}

__device__ __forceinline__ void rot2quat(const float* R, float* q) {
  float m00=R[0], m01=R[1], m02=R[2];
  float m10=R[3], m11=R[4], m12=R[5];
  float m20=R[6], m21=R[7], m22=R[8];
  float t0 = 1.f + m00 + m11 + m22;
  float t1 = 1.f + m00 - m11 - m22;
  float t2 = 1.f - m00 + m11 - m22;
  float t3 = 1.f - m00 - m11 + m22;
  float s0 = 2.f * sqrtf(fmaxf(t0, 1e-8f));
  float s1 = 2.f * sqrtf(fmaxf(t1, 1e-8f));
  float s2 = 2.f * sqrtf(fmaxf(t2, 1e-8f));
  float s3 = 2.f * sqrtf(fmaxf(t3, 1e-8f));
  float c[4][4] = {
    {0.25f*s0, (m21-m12)/s0, (m02-m20)/s0, (m10-m01)/s0},
    {(m21-m12)/s1, 0.25f*s1, (m01+m10)/s1, (m02+m20)/s1},
    {(m02-m20)/s2, (m01+m10)/s2, 0.25f*s2, (m12+m21)/s2},
    {(m10-m01)/s3, (m02+m20)/s3, (m12+m21)/s3, 0.25f*s3}};
  int idx = 0; float best = t0;
  if (t1 > best) { best = t1; idx = 1; }
  if (t2 > best) { best = t2; idx = 2; }
  if (t3 > best) { best = t3; idx = 3; }
  float n = 0.f;
  for (int k = 0; k < 4; ++k) n += c[idx][k] * c[idx][k];
  n = rsqrtf(n);
  for (int k = 0; k < 4; ++k) q[k] = c[idx][k] * n;
}

__global__ void k_frames_init(const float* fn, float* R, float* t, long long rows) {
  long long r = (long long)blockIdx.x * blockDim.x + threadIdx.x;
  if (r >= rows) return;
  const float* f = fn + r * 7;
  float n = sqrtf(f[0]*f[0] + f[1]*f[1] + f[2]*f[2] + f[3]*f[3]);
  n = fmaxf(n, 1e-8f);
  quat2rot(f[0]/n, f[1]/n, f[2]/n, f[3]/n, R + r * 9);
  t[r*3+0] = f[4] * 0.1f; t[r*3+1] = f[5] * 0.1f; t[r*3+2] = f[6] * 0.1f;
}

__global__ void k_frames_update(const float* upd, const float* fmask, float* R, float* t,
                                float* fout, long long rows) {
  long long r = (long long)blockIdx.x * blockDim.x + threadIdx.x;
  if (r >= rows) return;
  const float* u = upd + r * 6;
  float qn = rsqrtf(1.f + u[0]*u[0] + u[1]*u[1] + u[2]*u[2]);
  float Ru[9]; quat2rot(qn, u[0]*qn, u[1]*qn, u[2]*qn, Ru);
  float Ro[9];
  for (int k2 = 0; k2 < 9; ++k2) Ro[k2] = R[r*9 + k2];
  float Rn[9];
  for (int i2 = 0; i2 < 3; ++i2)
    for (int j2 = 0; j2 < 3; ++j2)
      Rn[i2*3+j2] = Ro[i2*3+0]*Ru[0*3+j2] + Ro[i2*3+1]*Ru[1*3+j2] + Ro[i2*3+2]*Ru[2*3+j2];
  float t0o = t[r*3+0], t1o = t[r*3+1], t2o = t[r*3+2];
  float tn0 = t0o + Ro[0]*u[3] + Ro[1]*u[4] + Ro[2]*u[5];
  float tn1 = t1o + Ro[3]*u[3] + Ro[4]*u[4] + Ro[5]*u[5];
  float tn2 = t2o + Ro[6]*u[3] + Ro[7]*u[4] + Ro[8]*u[5];
  float m = fmask[r];
  for (int k2 = 0; k2 < 9; ++k2) { Ro[k2] = m*Rn[k2] + (1.f-m)*Ro[k2]; R[r*9+k2] = Ro[k2]; }
  float tb0 = m*tn0 + (1.f-m)*t0o, tb1 = m*tn1 + (1.f-m)*t1o, tb2 = m*tn2 + (1.f-m)*t2o;
  t[r*3+0] = tb0; t[r*3+1] = tb1; t[r*3+2] = tb2;
  float q[4]; rot2quat(Ro, q);
  float* fo = fout + r * 7;
  fo[0]=q[0]; fo[1]=q[1]; fo[2]=q[2]; fo[3]=q[3];
  fo[4]=tb0*10.f; fo[5]=tb1*10.f; fo[6]=tb2*10.f;
}

__global__ void k_angle_final(const float* a2, float* unnorm, float* ang, long long rows) {
  long long r = (long long)blockIdx.x * blockDim.x + threadIdx.x;
  if (r >= rows) return;
  float x = a2[r*2], y = a2[r*2+1];
  unnorm[r*2] = x; unnorm[r*2+1] = y;
  float n = sqrtf(fmaxf(x*x + y*y, 1e-8f));
  ang[r*2] = x/n; ang[r*2+1] = y/n;
}

// ---------------------------------------------------------------- host: params
struct Lin { const float* w; const float* b; };
struct Lnp { const float* g; const float* b; };
struct BlockP {
  Lin bb;
  Lin e_init, e_trunk0, e_trunk1, e_final; Lnp e_ln;
  Lin i_q, i_kv, i_qp, i_kvp, i_b, i_out; const float* head_w;
  Lnp ipa_ln;
  Lin skip;
  Lin t_qkv[2], t_out[2], t_ff1[2], t_ff2[2]; Lnp t_ln1[2], t_ln2[2];
  Lin tfmr_post;
  Lin nt_l[2][3]; Lnp nt_ln;
};
struct Params {
  BlockP blk[4];
  Lin a_in, a_init, a_out; Lin a_blk[2][2];
};

struct Cur {
  void* const* din; const int* sz; int n; int i; bool strict; bool ok;
  const float* take(long long e) {
    if (i >= n) { ok = false; return nullptr; }
    if (strict && sz[i] != (int)e) { ok = false; return nullptr; }
    return (const float*)din[i++];
  }
};

static bool parse_params(Cur c, Params& P, bool sorted) {
  auto lin = [&](int di, int dz) {
    Lin l;
    if (sorted) { l.b = c.take(dz); l.w = c.take((long long)di * dz); }
    else        { l.w = c.take((long long)di * dz); l.b = c.take(dz); }
    return l;
  };
  auto lnp = [&](int d) {
    Lnp l;
    if (sorted) { l.b = c.take(d); l.g = c.take(d); }
    else        { l.g = c.take(d); l.b = c.take(d); }
    return l;
  };
  if (sorted) {  // jax tree flatten: dict keys alphabetical at every level; 'angle' < 'blocks'
    for (int a = 0; a < 2; ++a) { P.a_blk[a][0] = lin(128,128); P.a_blk[a][1] = lin(128,128); }
    P.a_in = lin(256,128); P.a_init = lin(256,128); P.a_out = lin(128,2);
    for (int i = 0; i < 4; ++i) {
      BlockP& B = P.blk[i];
      B.bb = lin(256,6);
      if (i < 3) { B.e_final = lin(384,128); B.e_init = lin(256,128); B.e_ln = lnp(128);
                   B.e_trunk0 = lin(384,384); B.e_trunk1 = lin(384,384); }
      B.i_b = lin(128,12); B.head_w = c.take(12); B.i_kv = lin(256,384); B.i_kvp = lin(256,432);
      B.i_out = lin(2112,256); B.i_q = lin(256,192); B.i_qp = lin(256,144);
      B.ipa_ln = lnp(256);
      for (int l = 0; l < 2; ++l) { B.nt_l[l][0]=lin(256,256); B.nt_l[l][1]=lin(256,256); B.nt_l[l][2]=lin(256,256); }
      B.nt_ln = lnp(256);
      B.skip = lin(256,64);
      for (int l = 0; l < 2; ++l) {
        B.t_ff1[l]=lin(320,320); B.t_ff2[l]=lin(320,320);
        B.t_ln1[l]=lnp(320); B.t_ln2[l]=lnp(320);
        B.t_out[l]=lin(320,320); B.t_qkv[l]=lin(320,960);
      }
      B.tfmr_post = lin(320,256);
    }
  } else {       // insertion-order recursion; 'blocks' first, lin = (w,b), lnp = (g,b)
    for (int i = 0; i < 4; ++i) {
      BlockP& B = P.blk[i];
      B.i_q = lin(256,192); B.i_kv = lin(256,384); B.i_qp = lin(256,144); B.i_kvp = lin(256,432);
      B.i_b = lin(128,12); B.head_w = c.take(12); B.i_out = lin(2112,256);
      B.ipa_ln = lnp(256);
      B.skip = lin(256,64);
      for (int l = 0; l < 2; ++l) {
        B.t_qkv[l]=lin(320,960); B.t_out[l]=lin(320,320);
        B.t_ff1[l]=lin(320,320); B.t_ff2[l]=lin(320,320);
        B.t_ln1[l]=lnp(320); B.t_ln2[l]=lnp(320);
      }
      B.tfmr_post = lin(320,256);
      for (int l = 0; l < 2; ++l) { B.nt_l[l][0]=lin(256,256); B.nt_l[l][1]=lin(256,256); B.nt_l[l][2]=lin(256,256); }
      B.nt_ln = lnp(256);
      B.bb = lin(256,6);
      if (i < 3) { B.e_init = lin(256,128); B.e_trunk0 = lin(384,384); B.e_trunk1 = lin(384,384);
                   B.e_final = lin(384,128); B.e_ln = lnp(128); }
    }
    P.a_in = lin(256,128); P.a_init = lin(256,128);
    for (int a = 0; a < 2; ++a) { P.a_blk[a][0] = lin(128,128); P.a_blk[a][1] = lin(128,128); }
    P.a_out = lin(128,2);
  }
  if (c.strict && c.i != c.n) return false;
  return c.ok;
}

// ---------------------------------------------------------------- host: gemm launcher
static void G(hipStream_t st, long long M, int N, int K, int batches, int hdim, int flags,
              const float* A, long long ao, long long ai, int lda,
              const float* Bm, long long bo, long long bi, int ldb,
              const float* bias,
              float* C, long long co, long long ci, int ldc,
              const float* Res = nullptr, int ldr = 0, const float* A2 = nullptr) {
  GemmP g;
  g.A = A; g.ao = ao; g.ai = ai; g.lda = lda; g.A2 = A2;
  g.Bm = Bm; g.bo = bo; g.bi = bi; g.ldb = ldb;
  g.bias = bias; g.Res = Res; g.ldr = ldr;
  g.C = C; g.co = co; g.ci = ci; g.ldc = ldc;
  g.M = (int)M; g.N = N; g.K = K; g.hdim = hdim; g.flags = flags;
  int nt = (N + 15) / 16; if (nt > 8) nt = 8;
  int gx = (N + 16 * nt - 1) / (16 * nt);
  dim3 gr((unsigned)gx, (unsigned)((M + 63) / 64), (unsigned)batches);
  int mode = (flags & GF_BT) ? 1 : ((flags & GF_RELUA) ? 2 : (A2 ? 3 : 0));
  switch (nt * 4 + mode) {
#define GC(NTV, MV) case (NTV*4+MV): k_gemm<NTV,MV><<<gr,128,0,st>>>(g); break;
    GC(1,0) GC(1,1) GC(1,2) GC(1,3)
    GC(2,0) GC(2,1) GC(2,2) GC(2,3)
    GC(3,0) GC(3,1) GC(3,2) GC(3,3)
    GC(4,0) GC(4,1) GC(4,2) GC(4,3)
    GC(5,0) GC(5,1) GC(5,2) GC(5,3)
    GC(6,0) GC(6,1) GC(6,2) GC(6,3)
    GC(7,0) GC(7,1) GC(7,2) GC(7,3)
    GC(8,0) GC(8,1) GC(8,2) GC(8,3)
#undef GC
    default: break;
  }
}

#define GR1(tot) dim3((unsigned)(((tot) + 255) / 256)), dim3(256), 0, stream

extern "C" void kernel_launch(void* const* d_in, const int* in_sizes, int n_in,
                              void* d_out, int out_size, void* d_ws, size_t ws_size,
                              hipStream_t stream) {
  (void)out_size; (void)ws_size;
  const int Nn = 384, CS = 256, CZ = 128, Hh = 12;
  const long long BN = 768, BNN = 294912;

  // ---- locate inputs / parameters (order-robust, verified via in_sizes) ----
  const float *seq_mask = nullptr, *frame_mask = nullptr, *frame_noised = nullptr,
              *seq_emb = nullptr, *pair_in = nullptr;
  Params P{};
  bool got = false;
  if (n_in >= 6 && in_sizes[0] == 768 && in_sizes[1] == 768 && in_sizes[2] == 5376 &&
      in_sizes[3] == 196608 && in_sizes[4] == 37748736) {
    for (int s = 0; s < 2 && !got; ++s) {
      Cur c{d_in, in_sizes, n_in, 5, true, true};
      Params Pt{};
      if (parse_params(c, Pt, s != 0)) { P = Pt; got = true; }
    }
    if (got) {
      seq_mask = (const float*)d_in[0]; frame_mask = (const float*)d_in[1];
      frame_noised = (const float*)d_in[2]; seq_emb = (const float*)d_in[3];
      pair_in = (const float*)d_in[4];
    }
  }
  if (!got && n_in >= 6 && in_sizes[0] == 768 && in_sizes[1] == 5376 &&
      in_sizes[2] == 37748736 && in_sizes[n_in-2] == 196608 && in_sizes[n_in-1] == 768) {
    for (int s = 1; s >= 0 && !got; --s) {
      Cur c{d_in, in_sizes, n_in - 2, 3, true, true};
      Params Pt{};
      if (parse_params(c, Pt, s != 0)) { P = Pt; got = true; }
    }
    if (got) {
      frame_mask = (const float*)d_in[0]; frame_noised = (const float*)d_in[1];
      pair_in = (const float*)d_in[2]; seq_emb = (const float*)d_in[n_in-2];
      seq_mask = (const float*)d_in[n_in-1];
    }
  }
  if (!got) {  // blind fallback: top insertion order + insertion-order params
    Cur c{d_in, in_sizes, n_in, 5, false, true};
    parse_params(c, P, false);
    seq_mask = (const float*)d_in[0]; frame_mask = (const float*)d_in[1];
    frame_noised = (const float*)d_in[2]; seq_emb = (const float*)d_in[3];
    pair_in = (const float*)d_in[4];
  }

  // ---- workspace carve ----
  char* wp = (char*)d_ws; size_t off = 0;
  auto alloc = [&](size_t bytes) -> float* {
    float* p = (float*)(wp + off); off = (off + bytes + 255) & ~(size_t)255; return p;
  };
  float* E    = alloc((size_t)BNN * 384 * 4);
  float* H1   = alloc((size_t)BNN * 384 * 4);
  float* H2   = alloc((size_t)BNN * 384 * 4);
  float* pair = alloc((size_t)BNN * 128 * 4);
  float* qk   = alloc((size_t)2 * 12 * 384 * 384 * 4);   // logits -> attention in-place
  float* ptqk = alloc((size_t)2 * 12 * 384 * 384 * 4);
  float* bprj = alloc((size_t)BNN * 12 * 4);
  float* att  = alloc((size_t)2 * 4 * 384 * 384 * 4);
  float* cat  = alloc((size_t)BN * 2112 * 4);
  float* sbuf = alloc((size_t)BN * 256 * 4);
  float* sini = alloc((size_t)BN * 256 * 4);
  float* ipab = alloc((size_t)BN * 256 * 4);
  float* tin  = alloc((size_t)BN * 320 * 4);
  float* tmpa = alloc((size_t)BN * 320 * 4);
  float* tmpb = alloc((size_t)BN * 320 * 4);
  float* obuf = alloc((size_t)BN * 320 * 4);
  float* qkvb = alloc((size_t)BN * 960 * 4);
  float* qbuf = alloc((size_t)BN * 192 * 4);
  float* kvbf = alloc((size_t)BN * 384 * 4);
  float* qpl  = alloc((size_t)BN * 144 * 4);
  float* kvpl = alloc((size_t)BN * 432 * 4);
  float* qpts = alloc((size_t)BN * 144 * 4);
  float* kvpt = alloc((size_t)BN * 432 * 4);
  float* qsq  = alloc((size_t)BN * 12 * 4);
  float* ksq  = alloc((size_t)BN * 12 * 4);
  float* optb = alloc((size_t)2 * 12 * 384 * 24 * 4);
  float* nbv  = alloc((size_t)BN * 128 * 4);
  float* h1n  = alloc((size_t)BN * 256 * 4);
  float* h2n  = alloc((size_t)BN * 256 * 4);
  float* angb = alloc((size_t)BN * 128 * 4);
  float* anghb= alloc((size_t)BN * 128 * 4);
  float* aoutb= alloc((size_t)BN * 2 * 4);
  float* Rb   = alloc((size_t)BN * 9 * 4);
  float* tb   = alloc((size_t)BN * 3 * 4);
  float* updb = alloc((size_t)BN * 6 * 4);

  float* out        = (float*)d_out;
  float* out_frames = out;
  float* out_s      = out + (long long)4 * BN * 7;
  float* out_pair   = out_s + BN * 256;
  float* out_unnorm = out_pair + (long long)BNN * 128;
  float* out_ang    = out_unnorm + BN * 2;

  hipStream_t st = stream;

  // ---- init ----
  k_maskmul<<<GR1(BN * CS)>>>(sini, seq_emb, seq_mask, CS, BN * CS);
  k_maskmul<<<GR1(BN * CS)>>>(sbuf, seq_emb, seq_mask, CS, BN * CS);
  hipMemcpyAsync(pair, pair_in, (size_t)BNN * CZ * 4, hipMemcpyDeviceToDevice, st);
  k_frames_init<<<GR1(BN)>>>(frame_noised, Rb, tb, BN);

  for (int i = 0; i < 4; ++i) {
    BlockP& Kb = P.blk[i];

    // ------------- IPA -------------
    G(st, BN,192,CS, 1,1,0, sbuf,0,0,CS, Kb.i_q.w,0,0,192,  Kb.i_q.b,  qbuf,0,0,192);
    G(st, BN,384,CS, 1,1,0, sbuf,0,0,CS, Kb.i_kv.w,0,0,384, Kb.i_kv.b, kvbf,0,0,384);
    G(st, BN,144,CS, 1,1,0, sbuf,0,0,CS, Kb.i_qp.w,0,0,144, Kb.i_qp.b, qpl,0,0,144);
    G(st, BN,432,CS, 1,1,0, sbuf,0,0,CS, Kb.i_kvp.w,0,0,432,Kb.i_kvp.b,kvpl,0,0,432);
    G(st, BNN,12,CZ, 1,1,0, pair,0,0,CZ, Kb.i_b.w,0,0,12,   Kb.i_b.b,  bprj,0,0,12);
    k_rigid<<<GR1(BN * 48)>>>(qpts, qpl, Rb, tb, 48, BN * 48);
    k_rigid<<<GR1(BN * 144)>>>(kvpt, kvpl, Rb, tb, 144, BN * 144);
    k_sumsq<<<GR1(BN * Hh)>>>(qsq, qpts, 12, 12, BN * Hh);
    k_sumsq<<<GR1(BN * Hh)>>>(ksq, kvpt, 36, 12, BN * Hh);
    // qk: batched per (b,h), NT GEMM, K = C_HID = 16
    G(st, 384,384,16, 24,12, GF_BT, qbuf,73728,16,192, kvbf,147456,32,384, nullptr,
      qk,1769472,147456,384);
    // point-dot: batched per (b,h), NT GEMM, K = PQ*3 = 12
    G(st, 384,384,12, 24,12, GF_BT, qpts,55296,12,144, kvpt,165888,36,432, nullptr,
      ptqk,1769472,147456,384);
    k_softmax_ipa<<<dim3(384,12,2),128,0,st>>>(qk, ptqk, bprj, qsq, ksq,
                                               Kb.head_w, seq_mask, Nn, Hh);
    // o = a @ v  -> cat[:, h*16 : h*16+16]
    G(st, 384,16,384, 24,12, 0, qk,1769472,147456,384, kvbf+16,147456,32,384, nullptr,
      cat,811008,16,2112);
    // o_pt(world) = a @ v_pts  -> optb[b,h,i,24]
    G(st, 384,24,384, 24,12, 0, qk,1769472,147456,384, kvpt+12,165888,36,432, nullptr,
      optb,110592,9216,24);
    // o_pair: batched per (b,i), (12 x 384) @ (384 x 128) -> cat[:, 576:2112]
    G(st, 12,128,384, 768,384, 0, qk,1769472,384,147456, pair,18874368,49152,128, nullptr,
      cat+576,811008,2112,128);
    k_ipa_finalize<<<GR1(BN * Hh * 8)>>>(cat, optb, Rb, tb, Nn, Hh, 8, BN * Hh * 8);
    G(st, BN,CS,2112, 1,1,0, cat,0,0,2112, Kb.i_out.w,0,0,CS, Kb.i_out.b, ipab,0,0,CS);
    k_ln<<<dim3((unsigned)BN),128,0,st>>>(ipab, sbuf, Kb.ipa_ln.g, Kb.ipa_ln.b,
                                          seq_mask, 1, 0, Nn, sbuf, CS);

    // ------------- transformer -------------
    k_copy2d<<<GR1(BN * CS)>>>(tin, 320, sbuf, CS, CS, BN * CS);
    G(st, BN,64,CS, 1,1,0, sini,0,0,CS, Kb.skip.w,0,0,64, Kb.skip.b, tin+256,0,0,320);
    for (int l = 0; l < 2; ++l) {
      G(st, BN,960,320, 1,1,0, tin,0,0,320, Kb.t_qkv[l].w,0,0,960, Kb.t_qkv[l].b, qkvb,0,0,960);
      G(st, 384,384,80, 8,4, GF_BT, qkvb,368640,80,960, qkvb+320,368640,80,960, nullptr,
        att,589824,147456,384);
      k_softmax_tfmr<<<dim3(384,4,2),128,0,st>>>(att, seq_mask, Nn, 4);
      G(st, 384,80,384, 8,4, 0, att,589824,147456,384, qkvb+640,368640,80,960, nullptr,
        obuf,122880,80,320);
      G(st, BN,320,320, 1,1,0, obuf,0,0,320, Kb.t_out[l].w,0,0,320, Kb.t_out[l].b,
        tmpa,0,0,320, tin,320);
      k_ln<<<dim3((unsigned)BN),128,0,st>>>(tmpa, nullptr, Kb.t_ln1[l].g, Kb.t_ln1[l].b,
                                            nullptr, 0, 0, Nn, tin, 320);
      G(st, BN,320,320, 1,1,GF_RELU, tin,0,0,320, Kb.t_ff1[l].w,0,0,320, Kb.t_ff1[l].b,
        tmpb,0,0,320);
      G(st, BN,320,320, 1,1,0, tmpb,0,0,320, Kb.t_ff2[l].w,0,0,320, Kb.t_ff2[l].b,
        tmpa,0,0,320, tin,320);
      k_ln<<<dim3((unsigned)BN),128,0,st>>>(tmpa, nullptr, Kb.t_ln2[l].g, Kb.t_ln2[l].b,
                                            nullptr, 0, 0, Nn, tin, 320);
    }
    G(st, BN,CS,320, 1,1,0, tin,0,0,320, Kb.tfmr_post.w,0,0,CS, Kb.tfmr_post.b,
      sbuf,0,0,CS, sbuf,CS);

    // ------------- node transition -------------
    for (int l = 0; l < 2; ++l) {
      G(st, BN,CS,CS, 1,1,GF_RELU, sbuf,0,0,CS, Kb.nt_l[l][0].w,0,0,CS, Kb.nt_l[l][0].b, h1n,0,0,CS);
      G(st, BN,CS,CS, 1,1,GF_RELU, h1n,0,0,CS,  Kb.nt_l[l][1].w,0,0,CS, Kb.nt_l[l][1].b, h2n,0,0,CS);
      G(st, BN,CS,CS, 1,1,0,       h2n,0,0,CS,  Kb.nt_l[l][2].w,0,0,CS, Kb.nt_l[l][2].b,
        sbuf,0,0,CS, sbuf,CS);
    }
    k_ln<<<dim3((unsigned)BN),128,0,st>>>(sbuf, nullptr, Kb.nt_ln.g, Kb.nt_ln.b,
                                          seq_mask, 0, 1, Nn, sbuf, CS);

    // ------------- frame update -------------
    G(st, BN,6,CS, 1,1,0, sbuf,0,0,CS, Kb.bb.w,0,0,6, Kb.bb.b, updb,0,0,6);
    k_frames_update<<<GR1(BN)>>>(updb, frame_mask, Rb, tb, out_frames + (long long)i * BN * 7, BN);

    // ------------- edge transition -------------
    if (i < 3) {
      G(st, BN,128,CS, 1,1,0, sbuf,0,0,CS, Kb.e_init.w,0,0,128, Kb.e_init.b, nbv,0,0,128);
      k_edge_concat<<<GR1(BNN * 384)>>>(E, pair, nbv, Nn, BNN * 384);
      G(st, BNN,384,384, 1,1,GF_RELU, E,0,0,384,  Kb.e_trunk0.w,0,0,384, Kb.e_trunk0.b, H1,0,0,384);
      G(st, BNN,384,384, 1,1,GF_RELU, H1,0,0,384, Kb.e_trunk1.w,0,0,384, Kb.e_trunk1.b, H2,0,0,384);
      // final( h2 + e ): dual-A fused add
      G(st, BNN,128,384, 1,1,0, H2,0,0,384, Kb.e_final.w,0,0,128, Kb.e_final.b,
        H1,0,0,128, nullptr,0, E);
      k_ln<<<dim3((unsigned)BNN),128,0,st>>>(H1, nullptr, Kb.e_ln.g, Kb.e_ln.b,
                                             seq_mask, 0, 2, Nn, pair, CZ);
    }
  }

  // ------------- angle resnet -------------
  G(st, BN,128,CS, 1,1,GF_RELUA, sbuf,0,0,CS, P.a_in.w,0,0,128, P.a_in.b, angb,0,0,128);
  G(st, BN,128,CS, 1,1,GF_RELUA, sini,0,0,CS, P.a_init.w,0,0,128, P.a_init.b,
    angb,0,0,128, angb,128);
  for (int a = 0; a < 2; ++a) {
    G(st, BN,128,128, 1,1,GF_RELUA, angb,0,0,128,  P.a_blk[a][0].w,0,0,128, P.a_blk[a][0].b,
      anghb,0,0,128);
    G(st, BN,128,128, 1,1,GF_RELUA, anghb,0,0,128, P.a_blk[a][1].w,0,0,128, P.a_blk[a][1].b,
      angb,0,0,128, angb,128);
  }
  G(st, BN,2,128, 1,1,GF_RELUA, angb,0,0,128, P.a_out.w,0,0,2, P.a_out.b, aoutb,0,0,2);
  k_angle_final<<<GR1(BN)>>>(aoutb, out_unnorm, out_ang, BN);

  hipMemcpyAsync(out_s, sbuf, (size_t)BN * CS * 4, hipMemcpyDeviceToDevice, st);
  hipMemcpyAsync(out_pair, pair, (size_t)BNN * CZ * 4, hipMemcpyDeviceToDevice, st);
}